// HungarianMatcher_7267084665334
// MI455X (gfx1250) — compile-verified
//
#include <hip/hip_runtime.h>
#include <hip/hip_bf16.h>

typedef float v2f __attribute__((ext_vector_type(2)));
typedef float v8f __attribute__((ext_vector_type(8)));

constexpr int kRowsPerBlock = 16;   // query rows per block
constexpr int kNcls         = 80;
constexpr int kWStride      = 84;   // padded W row stride (bank-conflict free A loads)
constexpr int kMTot         = 1024; // bs*T target columns
constexpr int kQTot         = 16000;// bs*Q rows

__global__ __launch_bounds__(256)
void matcher_cost_kernel(const float* __restrict__ pred_cls,
                         const float* __restrict__ pred_box,
                         const int*   __restrict__ tgt_ids,
                         const float* __restrict__ tgt_bbox,
                         float* __restrict__ out)
{
    __shared__ float  W_lds[kRowsPerBlock * kWStride]; // focal cls weight W = pos_w - neg_w
    __shared__ float4 box_lds[kMTot];                  // target boxes xyxy
    __shared__ int    ids_lds[kMTot];                  // target class ids
    __shared__ float4 qb_lds[kRowsPerBlock];           // this block's query boxes

    const int tid  = threadIdx.x;
    const int row0 = blockIdx.x * kRowsPerBlock;

    // ---------------- Phase 1: stage targets + build focal weight matrix ----------------
    for (int i = tid; i < kMTot; i += 256) {
        box_lds[i] = ((const float4*)tgt_bbox)[i];
        ids_lds[i] = tgt_ids[i];
    }
    if (tid < kRowsPerBlock)
        qb_lds[tid] = ((const float4*)pred_box)[row0 + tid];

    for (int i = tid; i < kRowsPerBlock * kNcls; i += 256) {
        int r = i / kNcls;
        int c = i - r * kNcls;
        float x   = pred_cls[(row0 + r) * kNcls + c];
        float p   = __builtin_amdgcn_rcpf(1.0f + __expf(-x)); // sigmoid
        float omp = 1.0f - p;
        float posw = 0.25f * omp * omp * (-__logf(p   + 1e-8f));
        float negw = 0.75f * p   * p   * (-__logf(omp + 1e-8f));
        W_lds[r * kWStride + c] = posw - negw;                // cost_cls coefficient
    }
    __syncthreads();

    // ---------------- Phase 2: per-wave 16x16 tiles, GIoU in VALU + cls via WMMA ----------------
    const int wave  = tid >> 5;
    const int lane  = tid & 31;
    const int lhalf = lane >> 4;   // 0: lanes 0-15, 1: lanes 16-31
    const int lmod  = lane & 15;

    // Uniform output base for this block's 16 contiguous rows (64 KB span):
    // stores become saddr-form global_store_b32 with the row step in the imm offset.
    float* __restrict__ outb = out + (size_t)row0 * kMTot;

    // A fragments for V_WMMA_F32_16X16X4_F32: element k-index = 4*chunk + vgpr + 2*lhalf
    v2f afrag[20];
#pragma unroll
    for (int c = 0; c < 20; ++c) {
        int kk = 4 * c + 2 * lhalf;
        afrag[c].x = W_lds[lmod * kWStride + kk];
        afrag[c].y = W_lds[lmod * kWStride + kk + 1];
    }

    // Row boxes in registers: vgpr v covers row (v + 8*lhalf) of the tile
    float4 qb[8];
    float  ar1[8];
#pragma unroll
    for (int v = 0; v < 8; ++v) {
        qb[v]  = qb_lds[v + 8 * lhalf];
        ar1[v] = (qb[v].z - qb[v].x) * (qb[v].w - qb[v].y);
    }

    const int rowoff = (lhalf * 8) * kMTot;    // per-lane-half row offset (elements)

    for (int t = 0; t < 8; ++t) {
        const int mbase = (wave * 8 + t) * 16;
        const int mcol  = mbase + lmod;          // lanes 16-31 broadcast-read same addrs
        const float4 b2 = box_lds[mcol];
        const int   id2 = ids_lds[mcol];
        const float a2  = (b2.z - b2.x) * (b2.w - b2.y);

        // Seed accumulator with the regression part:  1 - inter/union - union/area_e
        // (C = cls - GIoU = cls + 1 - iou - union/area_e)
        v8f acc;
#pragma unroll
        for (int v = 0; v < 8; ++v) {
            const float4 p = qb[v];
            float ltx = fmaxf(p.x, b2.x), lty = fmaxf(p.y, b2.y);
            float rbx = fminf(p.z, b2.z), rby = fminf(p.w, b2.w);
            float wx = fmaxf(rbx - ltx, 0.0f);
            float wy = fmaxf(rby - lty, 0.0f);
            float inter = wx * wy;
            float uni   = ar1[v] + a2 - inter;
            float lex = fminf(p.x, b2.x), ley = fminf(p.y, b2.y);
            float rex = fmaxf(p.z, b2.z), rey = fmaxf(p.w, b2.w);
            float ae  = (rex - lex) * (rey - ley);
            acc[v] = 1.0f - inter * __builtin_amdgcn_rcpf(uni)
                          - uni   * __builtin_amdgcn_rcpf(ae);
        }

        // cls gather as exact one-hot matmul: D = W(16x80) x OneHot(80x16) + acc
#pragma unroll
        for (int c = 0; c < 20; ++c) {
            int kk = 4 * c + 2 * lhalf;          // B element k-index = 4*chunk + vgpr + 2*lhalf
            v2f bb;
            bb.x = (id2 == kk)     ? 1.0f : 0.0f;
            bb.y = (id2 == kk + 1) ? 1.0f : 0.0f;
            acc = __builtin_amdgcn_wmma_f32_16x16x4_f32(
                      /*neg_a=*/false, afrag[c],
                      /*neg_b=*/false, bb,
                      /*c_mod=*/(short)0, acc,
                      /*reuse_a=*/false, /*reuse_b=*/false);
        }

        // Store: vgpr v -> row (v + 8*lhalf); one 32-bit lane offset, row step is an
        // immediate multiple of 4096 B; streamed output -> non-temporal.
        const int off0 = rowoff + mcol;
#pragma unroll
        for (int v = 0; v < 8; ++v) {
            __builtin_nontemporal_store(acc[v], &outb[off0 + v * kMTot]);
        }
    }
}

extern "C" void kernel_launch(void* const* d_in, const int* in_sizes, int n_in,
                              void* d_out, int out_size, void* d_ws, size_t ws_size,
                              hipStream_t stream) {
    (void)in_sizes; (void)n_in; (void)out_size; (void)d_ws; (void)ws_size;
    const float* pred_cls = (const float*)d_in[0];
    const float* pred_box = (const float*)d_in[1];
    const int*   tgt_ids  = (const int*)d_in[2];
    const float* tgt_bbox = (const float*)d_in[3];
    float* out = (float*)d_out;

    dim3 grid(kQTot / kRowsPerBlock); // 1000 blocks, 16 rows each
    matcher_cost_kernel<<<grid, 256, 0, stream>>>(pred_cls, pred_box, tgt_ids, tgt_bbox, out);
}